// SparseModel_70291434766874
// MI455X (gfx1250) — compile-verified
//
#include <hip/hip_runtime.h>

typedef __attribute__((ext_vector_type(2))) float v2f;
typedef __attribute__((ext_vector_type(8))) float v8f;

#define IN_DIM  64
#define HIDDEN  128
#define OUT_DIM 32
#define MT      64    // batch rows per workgroup
#define LSTRIDE 132   // padded LDS row stride in floats (132 % 64 == 4 -> conflict-free frag loads)

__device__ __forceinline__ float swish_act(float v) {
    return v * (1.0f / (1.0f + __expf(-v)));
}

// One fused layer: H_out = swish( H_in(in LDS, MT x K) * W(K x HIDDEN, global) + bias )
// Accumulators live in VGPRs across the barrier, so LDS is overwritten in place.
template <int K>
__device__ __forceinline__ void gemm_layer(float* lds,
                                           const float* __restrict__ W,
                                           const float* __restrict__ bias,
                                           int col, int half, int r16) {
    v8f c[4] = {};   // 4 M-tiles (64 rows) x this wave's 16-column N-tile

    #pragma unroll
    for (int k0 = 0; k0 < K; k0 += 4) {
        // B fragment: 4x16 tile of W. lane<16 -> rows k0,k0+1 ; lane>=16 -> rows k0+2,k0+3
        const float* wp = W + (size_t)(k0 + 2 * half) * HIDDEN + col;
        v2f bfrag;
        bfrag.x = wp[0];
        bfrag.y = wp[HIDDEN];

        #pragma unroll
        for (int m = 0; m < 4; ++m) {
            // A fragment: rows 16m..16m+15, lane<16 -> k0,k0+1 ; lane>=16 -> k0+2,k0+3
            const int row = 16 * m + r16;
            const float2 af = *(const float2*)(&lds[row * LSTRIDE + k0 + 2 * half]);
            v2f afrag;
            afrag.x = af.x;
            afrag.y = af.y;
            c[m] = __builtin_amdgcn_wmma_f32_16x16x4_f32(
                false, afrag, false, bfrag, (short)0, c[m], false, false);
        }
    }

    __syncthreads();   // all waves done reading H_in from LDS

    const float bv = bias[col];
    #pragma unroll
    for (int m = 0; m < 4; ++m) {
        #pragma unroll
        for (int rr = 0; rr < 8; ++rr) {
            // C/D layout: VGPR rr holds M=rr (lanes 0-15) / M=rr+8 (lanes 16-31); N = lane&15
            const int row = 16 * m + rr + 8 * half;
            lds[row * LSTRIDE + col] = swish_act(c[m][rr] + bv);
        }
    }
    __syncthreads();   // H_out visible to all waves
}

__global__ void __launch_bounds__(256)
grouped_mlp_kernel(const float* __restrict__ x,
                   const float* __restrict__ W1,
                   const float* __restrict__ b1,
                   const float* __restrict__ Wh,
                   const float* __restrict__ bh,
                   const float* __restrict__ Wo,
                   const float* __restrict__ bo,
                   float* __restrict__ out,
                   int N) {
    __shared__ float lds[MT * LSTRIDE];   // 33 KB activation buffer (reused across layers)

    const int o    = blockIdx.y;          // subnet
    const int n0   = blockIdx.x * MT;     // batch row base
    const int t    = threadIdx.x;
    const int wave = t >> 5;
    const int lane = t & 31;
    const int half = lane >> 4;
    const int r16  = lane & 15;
    const int col  = wave * 16 + r16;     // this lane's hidden column (wave owns 16 cols)

    // ---- stage X tile [MT x IN_DIM] into LDS (float4, coalesced) ----
    #pragma unroll
    for (int i = 0; i < (MT * IN_DIM / 4) / 256; ++i) {
        const int linear = t + 256 * i;       // float4 id: 16 per row
        const int row    = linear >> 4;
        const int c4     = linear & 15;
        const float4 v = *(const float4*)(x + (size_t)(n0 + row) * IN_DIM + c4 * 4);
        *(float4*)(&lds[row * LSTRIDE + c4 * 4]) = v;
    }
    __syncthreads();

    // ---- layer 1: 64 -> 128 ----
    gemm_layer<IN_DIM>(lds, W1 + (size_t)o * IN_DIM * HIDDEN, b1 + (size_t)o * HIDDEN,
                       col, half, r16);
    // ---- hidden layers: 128 -> 128 (N_LAYERS-1 = 2) ----
    gemm_layer<HIDDEN>(lds, Wh + ((size_t)0 * OUT_DIM + o) * HIDDEN * HIDDEN,
                       bh + ((size_t)0 * OUT_DIM + o) * HIDDEN, col, half, r16);
    gemm_layer<HIDDEN>(lds, Wh + ((size_t)1 * OUT_DIM + o) * HIDDEN * HIDDEN,
                       bh + ((size_t)1 * OUT_DIM + o) * HIDDEN, col, half, r16);

    // ---- head: out[n, o] = dot(h[n,:], Wo[o,:]) + bo[o] ----
    {
        const int row  = t >> 2;        // 4 lanes per batch row
        const int part = t & 3;
        const float* wo = Wo + (size_t)o * HIDDEN;
        float s = 0.0f;
        #pragma unroll
        for (int k = 0; k < HIDDEN / 4; ++k) {
            const int kk = part * (HIDDEN / 4) + k;
            s += lds[row * LSTRIDE + kk] * wo[kk];
        }
        s += __shfl_xor(s, 1, 32);
        s += __shfl_xor(s, 2, 32);
        if (part == 0) {
            out[(size_t)(n0 + row) * OUT_DIM + o] = s + bo[o];
        }
    }
}

extern "C" void kernel_launch(void* const* d_in, const int* in_sizes, int n_in,
                              void* d_out, int out_size, void* d_ws, size_t ws_size,
                              hipStream_t stream) {
    const float* x  = (const float*)d_in[0];
    const float* W1 = (const float*)d_in[1];
    const float* b1 = (const float*)d_in[2];
    const float* Wh = (const float*)d_in[3];
    const float* bh = (const float*)d_in[4];
    const float* Wo = (const float*)d_in[5];
    const float* bo = (const float*)d_in[6];
    float* out = (float*)d_out;

    const int N = in_sizes[0] / IN_DIM;   // 32768
    dim3 grid(N / MT, OUT_DIM);           // (512, 32) workgroups
    grouped_mlp_kernel<<<grid, 256, 0, stream>>>(x, W1, b1, Wh, bh, Wo, bo, out, N);
}